// GNNLayer_137438954334
// MI455X (gfx1250) — compile-verified
//
#include <hip/hip_runtime.h>
#include <hip/hip_bf16.h>

// ---------------------------------------------------------------------------
// GDN GraphLayer (GAT-like) for MI455X / gfx1250, all fp32.
//   h = x @ W^T          -> WMMA f32 16x16x4 (exact fp32 matrix path)
//   s_i/s_j              -> per-node dots
//   edge softmax         -> atomic-max / exp+atomic-add / normalize
//   message scatter      -> per-(edge,channel) atomicAdd (L2-resident h)
//   out = relu(acc+bias)
// ---------------------------------------------------------------------------

typedef __attribute__((ext_vector_type(2))) float v2f;
typedef __attribute__((ext_vector_type(8))) float v8f;

#define C64 64          // IN_C == OUT_C == D_EMB == 64
#define LDP 68          // LDS row stride (floats), pads 64-bank conflicts, 16B aligned

// ------------------------- h = x @ W^T  (WMMA fp32) -------------------------
// One wave computes a 16x64 tile of h. W staged block-wide in LDS, each
// wave's 16x64 x-tile staged in LDS. 16 K-steps of V_WMMA_F32_16X16X4_F32
// per column tile (4 col tiles) = 64 wmma per wave.
__global__ __launch_bounds__(256) void gemm_h_kernel(const float* __restrict__ x,
                                                     const float* __restrict__ W,
                                                     float* __restrict__ h,
                                                     int nrows) {
    __shared__ float Wl[C64 * LDP];           // W[o][k], padded
    __shared__ float Xl[8][16 * LDP];         // per-wave x tile, padded

    const int tid  = threadIdx.x;
    const int wave = tid >> 5;
    const int lane = tid & 31;
    const int lm   = lane & 15;
    const int hi   = lane >> 4;               // 0: K=0,1 half   1: K=2,3 half

    // Stage W (64x64 = 4096 floats) with coalesced float4 loads.
    for (int i = tid * 4; i < C64 * C64; i += 256 * 4) {
        const float4 w4 = *(const float4*)(W + i);
        const int r = i >> 6, c = i & 63;
        *(float4*)(&Wl[r * LDP + c]) = w4;
    }

    const int row0   = (blockIdx.x * 8 + wave) * 16;
    const bool active = (row0 + 16) <= nrows;  // N=50000 is a multiple of 16

    if (active) {
        // Stage this wave's 16x64 x tile (1024 floats, 8 float4 per lane).
        for (int i = lane * 4; i < 16 * C64; i += 32 * 4) {
            const int r = i >> 6, c = i & 63;
            const float4 v = *(const float4*)(x + (size_t)(row0 + r) * C64 + c);
            *(float4*)(&Xl[wave][r * LDP + c]) = v;
        }
    }
    __syncthreads();

    if (active) {
        v8f acc[4] = {};
        #pragma unroll
        for (int t = 0; t < 16; ++t) {
            const int k = t * 4 + hi * 2;
            // A fragment: 16x4 f32 (lane<16: M=lane,K=k..k+1; lane>=16: K=k+2..k+3)
            v2f a;
            a.x = Xl[wave][lm * LDP + k];
            a.y = Xl[wave][lm * LDP + k + 1];
            #pragma unroll
            for (int c = 0; c < 4; ++c) {
                const int col = c * 16 + lm;
                // B fragment: 4x16 f32 of W^T -> B[k][col] = W[col][k]
                v2f b;
                b.x = Wl[col * LDP + k];
                b.y = Wl[col * LDP + k + 1];
                acc[c] = __builtin_amdgcn_wmma_f32_16x16x4_f32(
                    /*neg_a=*/false, a, /*neg_b=*/false, b,
                    /*c_mod=*/(short)0, acc[c],
                    /*reuse_a=*/false, /*reuse_b=*/false);
            }
        }
        // C/D layout: VGPR v -> M = v + 16*hi/2 halves: lanes 0-15 M=v, lanes 16-31 M=v+8
        #pragma unroll
        for (int c = 0; c < 4; ++c) {
            #pragma unroll
            for (int v = 0; v < 8; ++v) {
                const int m = v + hi * 8;
                h[(size_t)(row0 + m) * C64 + c * 16 + lm] = acc[c][v];
            }
        }
    }
}

// ------------------------- per-node attention scores ------------------------
__global__ void scores_kernel(const float* __restrict__ h,
                              const float* __restrict__ emb,
                              const float* __restrict__ att_i,
                              const float* __restrict__ att_j,
                              const float* __restrict__ att_em_i,
                              const float* __restrict__ att_em_j,
                              float* __restrict__ s_i,
                              float* __restrict__ s_j, int n) {
    const int i = blockIdx.x * blockDim.x + threadIdx.x;
    if (i >= n) return;
    const float* hr = h + (size_t)i * C64;
    const float* er = emb + (size_t)i * C64;
    float si = 0.f, sj = 0.f;
    #pragma unroll 8
    for (int k = 0; k < C64; ++k) {
        const float hv = hr[k], ev = er[k];
        si += hv * att_i[k] + ev * att_em_i[k];
        sj += hv * att_j[k] + ev * att_em_j[k];
    }
    s_i[i] = si;
    s_j[i] = sj;
}

// ------------------------------ init scratch --------------------------------
__global__ void init_kernel(float* __restrict__ amax, float* __restrict__ denom,
                            float* __restrict__ accum, int n) {
    const int i = blockIdx.x * blockDim.x + threadIdx.x;
    if (i < n) { amax[i] = -__builtin_inff(); denom[i] = 0.f; }
    if (i < n * C64) accum[i] = 0.f;
}

// float atomic max via int/uint monotone bit trick (always compiles; lowers to
// global_atomic_max_i32 / global_atomic_min_u32)
__device__ __forceinline__ void atomicMaxF(float* addr, float v) {
    if (v >= 0.f) atomicMax((int*)addr, __float_as_int(v));
    else          atomicMin((unsigned int*)addr, (unsigned int)__float_as_int(v));
}

__device__ __forceinline__ void edge_ids(const int* __restrict__ ei, long e,
                                         int E, int& src, int& dst) {
    if (e < E) { src = ei[e]; dst = ei[(size_t)E + e]; }
    else       { src = dst = (int)(e - E); }
}

// ------------------- pass 1: logits + leaky relu + seg max ------------------
__global__ void edge_logits_kernel(const int* __restrict__ ei,
                                   const float* __restrict__ s_i,
                                   const float* __restrict__ s_j,
                                   float* __restrict__ alpha,
                                   float* __restrict__ amax, int E, int n) {
    const long e = (long)blockIdx.x * blockDim.x + threadIdx.x;
    if (e >= (long)E + n) return;
    int src, dst;
    edge_ids(ei, e, E, src, dst);
    float a = s_i[dst] + s_j[src];
    a = (a >= 0.f) ? a : 0.2f * a;   // NEG_SLOPE
    alpha[e] = a;
    atomicMaxF(amax + dst, a);
}

// ------------------- pass 2: exp + seg sum (denominator) --------------------
__global__ void edge_exp_kernel(const int* __restrict__ ei,
                                const float* __restrict__ amax,
                                float* __restrict__ alpha,   // in: logits, out: exp
                                float* __restrict__ denom, int E, int n) {
    const long e = (long)blockIdx.x * blockDim.x + threadIdx.x;
    if (e >= (long)E + n) return;
    int src, dst;
    edge_ids(ei, e, E, src, dst);
    const float ex = __expf(alpha[e] - amax[dst]);
    alpha[e] = ex;
    atomicAdd(denom + dst, ex);
}

// -------- pass 3: scatter messages; lane = channel for coalescing -----------
__global__ __launch_bounds__(256) void edge_scatter_kernel(const int* __restrict__ ei,
                                                           const float* __restrict__ h,
                                                           const float* __restrict__ ex,
                                                           const float* __restrict__ denom,
                                                           float* __restrict__ accum,
                                                           int E, int n) {
    const int c  = threadIdx.x & 63;          // channel
    const int eo = threadIdx.x >> 6;          // 4 edges per block
    const long e = (long)blockIdx.x * 4 + eo;
    if (e >= (long)E + n) return;
    int src, dst;
    edge_ids(ei, e, E, src, dst);
    const float coef = ex[e] / (denom[dst] + 1e-16f);
    atomicAdd(accum + (size_t)dst * C64 + c, h[(size_t)src * C64 + c] * coef);
}

// ------------------------------ bias + relu ---------------------------------
__global__ void finalize_kernel(const float* __restrict__ accum,
                                const float* __restrict__ bias,
                                float* __restrict__ out, int total) {
    const int i = blockIdx.x * blockDim.x + threadIdx.x;
    if (i >= total) return;
    const float v = accum[i] + bias[i & 63];
    out[i] = v > 0.f ? v : 0.f;
}

// ---------------------------------------------------------------------------
extern "C" void kernel_launch(void* const* d_in, const int* in_sizes, int n_in,
                              void* d_out, int out_size, void* d_ws, size_t ws_size,
                              hipStream_t stream) {
    const float* x        = (const float*)d_in[0];
    const int*   ei       = (const int*)  d_in[1];
    const float* emb      = (const float*)d_in[2];
    const float* W        = (const float*)d_in[3];
    const float* att_i    = (const float*)d_in[4];
    const float* att_j    = (const float*)d_in[5];
    const float* att_em_i = (const float*)d_in[6];
    const float* att_em_j = (const float*)d_in[7];
    const float* bias     = (const float*)d_in[8];
    float* out = (float*)d_out;

    const int N = in_sizes[0] / C64;      // 50000
    const int E = in_sizes[1] / 2;        // 800000
    const long T = (long)E + N;           // edges incl. self loops

    // workspace layout (floats)
    float* ws    = (float*)d_ws;
    float* h     = ws;                       // N*64
    float* accum = h     + (size_t)N * C64;  // N*64
    float* s_i   = accum + (size_t)N * C64;  // N
    float* s_j   = s_i   + N;                // N
    float* amax  = s_j   + N;                // N
    float* denom = amax  + N;                // N
    float* alpha = denom + N;                // E+N

    // 1) h = x @ W^T via WMMA fp32
    const int row_tiles = (N + 15) / 16;
    const int gemm_blocks = (row_tiles + 7) / 8;
    gemm_h_kernel<<<gemm_blocks, 256, 0, stream>>>(x, W, h, N);

    // 2) per-node scores
    scores_kernel<<<(N + 255) / 256, 256, 0, stream>>>(h, emb, att_i, att_j,
                                                       att_em_i, att_em_j,
                                                       s_i, s_j, N);

    // 3) init amax/denom/accum
    init_kernel<<<((int)((size_t)N * C64) + 255) / 256, 256, 0, stream>>>(amax, denom, accum, N);

    // 4) softmax passes over edges
    const int eb = (int)((T + 255) / 256);
    edge_logits_kernel<<<eb, 256, 0, stream>>>(ei, s_i, s_j, alpha, amax, E, N);
    edge_exp_kernel<<<eb, 256, 0, stream>>>(ei, amax, alpha, denom, E, N);

    // 5) message scatter (4 edges x 64 channels per block)
    const int sb = (int)((T + 3) / 4);
    edge_scatter_kernel<<<sb, 256, 0, stream>>>(ei, h, alpha, denom, accum, E, N);

    // 6) bias + relu -> d_out
    finalize_kernel<<<((int)((size_t)N * C64) + 255) / 256, 256, 0, stream>>>(accum, bias, out, N * C64);
}